// PERConv_11716670783823
// MI455X (gfx1250) — compile-verified
//
#include <hip/hip_runtime.h>
#include <hip/hip_bf16.h>

#define NNODES   100000
#define NEDGES   1600000
#define C        64
#define NGRAPH   64
#define NTILES   (NNODES / 16)            // 6250 exact
#define LSTR     68                       // LDS row stride in floats
#define NSCB     ((NNODES + 255) / 256)   // 391 scan blocks

typedef __attribute__((ext_vector_type(2))) float v2f;
typedef __attribute__((ext_vector_type(8))) float v8f;

// ---------------------------------------------------------------------------
// fp32 WMMA: D = A(16x4) * B(4x16) + C(16x16)
// A frag (lane l): M = l%16, holds K = {2*(l/16), 2*(l/16)+1}
// B frag (lane l): N = l%16, holds K = {2*(l/16), 2*(l/16)+1}
// C/D (lane l, reg r): M = r + 8*(l/16), N = l%16
// Weights live TRANSPOSED in LDS (Wt[col][k]) so both A and B fragments are
// single contiguous ds_load_b64's into adjacent VGPR pairs.
// ---------------------------------------------------------------------------
__device__ __forceinline__ v8f wmma_f32(v2f a, v2f b, v8f c) {
    return __builtin_amdgcn_wmma_f32_16x16x4_f32(false, a, false, b, (short)0, c,
                                                 false, false);
}

__device__ __forceinline__ int lower_bound_i(const int* __restrict__ arr, int n, int key) {
    int lo = 0, hi = n;
    while (lo < hi) {
        int mid = (lo + hi) >> 1;
        if (arr[mid] < key) lo = mid + 1; else hi = mid;
    }
    return lo;
}

// ---------------------------------------------------------------------------
// CSR build: zero -> count -> scan(3) -> fill
// ---------------------------------------------------------------------------
__global__ __launch_bounds__(256) void k_zero_i(int* __restrict__ a, int* __restrict__ b, int n) {
    int i = blockIdx.x * 256 + threadIdx.x;
    if (i < n) { a[i] = 0; b[i] = 0; }
}

__global__ __launch_bounds__(256) void k_count(const int* __restrict__ ei,
                                               int* __restrict__ cnt) {
    int e = blockIdx.x * 256 + threadIdx.x;
    if (e < NEDGES) atomicAdd(&cnt[ei[NEDGES + e]], 1);
}

__global__ __launch_bounds__(256) void k_scan1(const int* __restrict__ cnt,
                                               int* __restrict__ off,
                                               int* __restrict__ bsum) {
    __shared__ int sc[256];
    const int t = threadIdx.x;
    const int i = blockIdx.x * 256 + t;
    int v = (i < NNODES) ? cnt[i] : 0;
    sc[t] = v;
    __syncthreads();
    for (int d = 1; d < 256; d <<= 1) {
        int y = (t >= d) ? sc[t - d] : 0;
        __syncthreads();
        sc[t] += y;
        __syncthreads();
    }
    if (i < NNODES) off[i] = sc[t] - v;          // exclusive within block
    if (t == 255) bsum[blockIdx.x] = sc[255];
}

__global__ __launch_bounds__(512) void k_scan2(int* __restrict__ bsum) {
    __shared__ int sc[512];
    const int t = threadIdx.x;
    int v = (t < NSCB) ? bsum[t] : 0;
    sc[t] = v;
    __syncthreads();
    for (int d = 1; d < 512; d <<= 1) {
        int y = (t >= d) ? sc[t - d] : 0;
        __syncthreads();
        sc[t] += y;
        __syncthreads();
    }
    if (t < NSCB) bsum[t] = sc[t] - v;           // exclusive
}

__global__ __launch_bounds__(256) void k_scan3(int* __restrict__ off,
                                               const int* __restrict__ bsum) {
    int i = blockIdx.x * 256 + threadIdx.x;
    if (i < NNODES) off[i] += bsum[blockIdx.x];
}

__global__ __launch_bounds__(256) void k_fill(const int* __restrict__ ei,
                                              const int* __restrict__ off,
                                              int* __restrict__ cur,
                                              int* __restrict__ esrc) {
    int e = blockIdx.x * 256 + threadIdx.x;
    if (e >= NEDGES) return;
    const int src = ei[e];
    const int dst = ei[NEDGES + e];
    const int pos = off[dst] + atomicAdd(&cur[dst], 1);
    esrc[pos] = src;
}

// ---------------------------------------------------------------------------
// Kernel 1: per-node precompute via WMMA.
//   DstT[n] = x[n]@(W_i - W_d) - pos[n]@W_p + msg_b
//   SrcT[n] = x[n]@W_d        + pos[n]@W_p
// ---------------------------------------------------------------------------
__global__ __launch_bounds__(128) void k_node_pre(
    const float* __restrict__ x, const float* __restrict__ pos,
    const float* __restrict__ msg_W, const float* __restrict__ msg_b,
    float* __restrict__ DstT, float* __restrict__ SrcT)
{
    __shared__ float W1t[64 * LSTR];     // (W_i - W_d) transposed: [col][k]
    __shared__ float W2t[64 * LSTR];     // W_d transposed:         [col][k]
    __shared__ float Xs[4][16 * LSTR];
    __shared__ float Wp[3][64];
    __shared__ float Bb[64];

    const int tid  = threadIdx.x;
    const int wave = tid >> 5;
    const int lane = tid & 31;
    const int half = lane >> 4;
    const int mrow = lane & 15;
    const int tile = blockIdx.x * 4 + wave;

    // Stage weights transposed: 1024 float4 loads, scatter 4 scalars each
#pragma unroll
    for (int i = 0; i < 8; ++i) {
        int f   = i * 128 + tid;
        int row = f >> 4, c4 = (f & 15) * 4;      // row = k, c4 = col base
        float4 wi = *(const float4*)(msg_W + row * 64 + c4);
        float4 wd = *(const float4*)(msg_W + (row + 64) * 64 + c4);
        W1t[(c4 + 0) * LSTR + row] = wi.x - wd.x;
        W1t[(c4 + 1) * LSTR + row] = wi.y - wd.y;
        W1t[(c4 + 2) * LSTR + row] = wi.z - wd.z;
        W1t[(c4 + 3) * LSTR + row] = wi.w - wd.w;
        W2t[(c4 + 0) * LSTR + row] = wd.x;
        W2t[(c4 + 1) * LSTR + row] = wd.y;
        W2t[(c4 + 2) * LSTR + row] = wd.z;
        W2t[(c4 + 3) * LSTR + row] = wd.w;
    }
    if (tid < 64) {
        Wp[0][tid] = msg_W[128 * 64 + tid];
        Wp[1][tid] = msg_W[129 * 64 + tid];
        Wp[2][tid] = msg_W[130 * 64 + tid];
        Bb[tid]    = msg_b[tid];
    }
    if (tile < NTILES) {
        const int base = tile * 16;
#pragma unroll
        for (int i = 0; i < 8; ++i) {
            int f   = i * 32 + lane;
            int row = f >> 4, c4 = (f & 15) * 4;
            *(float4*)(&Xs[wave][row * LSTR + c4]) =
                *(const float4*)(x + (size_t)(base + row) * 64 + c4);
        }
    }
    __syncthreads();
    if (tile >= NTILES) return;

    v8f accD[4] = {}, accS[4] = {};
    for (int kk = 0; kk < 16; ++kk) {
        const int k = kk * 4 + half * 2;
        const v2f a = *(const v2f*)(&Xs[wave][mrow * LSTR + k]);
#pragma unroll
        for (int ct = 0; ct < 4; ++ct) {
            const int col = ct * 16 + mrow;
            const v2f b1 = *(const v2f*)(&W1t[col * LSTR + k]);
            const v2f b2 = *(const v2f*)(&W2t[col * LSTR + k]);
            accD[ct] = wmma_f32(a, b1, accD[ct]);
            accS[ct] = wmma_f32(a, b2, accS[ct]);
        }
    }

    const int base = tile * 16;
#pragma unroll
    for (int r = 0; r < 8; ++r) {
        const int row = base + r + 8 * half;
        const float p0 = pos[(size_t)row * 3 + 0];
        const float p1 = pos[(size_t)row * 3 + 1];
        const float p2 = pos[(size_t)row * 3 + 2];
#pragma unroll
        for (int ct = 0; ct < 4; ++ct) {
            const int col = ct * 16 + mrow;
            const float pw = p0 * Wp[0][col] + p1 * Wp[1][col] + p2 * Wp[2][col];
            DstT[(size_t)row * 64 + col] = accD[ct][r] - pw + Bb[col];
            SrcT[(size_t)row * 64 + col] = accS[ct][r] + pw;
        }
    }
}

// ---------------------------------------------------------------------------
// Kernel 2: gather-max per destination (CSR). One wave per dst node.
//   agg[dst] = DstT[dst] + max_{e in in(dst)} SrcT[src_e];  deg==0 -> 0
// ---------------------------------------------------------------------------
__global__ __launch_bounds__(256) void k_aggregate(
    const float* __restrict__ DstT, const float* __restrict__ SrcT,
    const int* __restrict__ esrc, const int* __restrict__ off,
    const int* __restrict__ cnt, float* __restrict__ agg)
{
    const int wave = threadIdx.x >> 5;
    const int lane = threadIdx.x & 31;
    const int dst  = blockIdx.x * 8 + wave;
    if (dst >= NNODES) return;

    const int start = off[dst];
    const int deg   = cnt[dst];
    const int c0    = lane * 2;

    float m0 = 0.0f, m1 = 0.0f;
    if (deg > 0) {
        const float ninf = __int_as_float(0xff800000);
        m0 = ninf; m1 = ninf;
        int i = 0;
        for (; i + 4 <= deg; i += 4) {               // 4-deep for load MLP
            const int s0 = esrc[start + i + 0];
            const int s1 = esrc[start + i + 1];
            const int s2 = esrc[start + i + 2];
            const int s3 = esrc[start + i + 3];
            const float2 a0 = *(const float2*)(SrcT + (size_t)s0 * 64 + c0);
            const float2 a1 = *(const float2*)(SrcT + (size_t)s1 * 64 + c0);
            const float2 a2 = *(const float2*)(SrcT + (size_t)s2 * 64 + c0);
            const float2 a3 = *(const float2*)(SrcT + (size_t)s3 * 64 + c0);
            m0 = fmaxf(fmaxf(fmaxf(m0, a0.x), a1.x), fmaxf(a2.x, a3.x));
            m1 = fmaxf(fmaxf(fmaxf(m1, a0.y), a1.y), fmaxf(a2.y, a3.y));
        }
        for (; i < deg; ++i) {
            const int s = esrc[start + i];
            const float2 a = *(const float2*)(SrcT + (size_t)s * 64 + c0);
            m0 = fmaxf(m0, a.x);
            m1 = fmaxf(m1, a.y);
        }
        const float2 d = *(const float2*)(DstT + (size_t)dst * 64 + c0);
        m0 += d.x; m1 += d.y;
    }
    float2* o = (float2*)(agg + (size_t)dst * 64 + c0);
    *o = make_float2(m0, m1);
}

// ---------------------------------------------------------------------------
// Kernel 3: per-graph GraphNorm stats -> per-(graph,channel) k,d:
//   out = relu(a*k + d),  k = gn_w*rsqrt(var+eps),  d = gn_b - mean*gn_s*k
//   var = E[a^2] + mean^2 * gn_s * (gn_s - 2)
// ---------------------------------------------------------------------------
__global__ __launch_bounds__(256) void k_graph_stats(
    const float* __restrict__ agg, const int* __restrict__ batch,
    const float* __restrict__ gn_w, const float* __restrict__ gn_b,
    const float* __restrict__ gn_s, float* __restrict__ Kb, float* __restrict__ Db)
{
    __shared__ int s_lo, s_hi;
    __shared__ float sS[4][64], sQ[4][64];
    const int g = blockIdx.x, tid = threadIdx.x;
    if (tid == 0)  s_lo = lower_bound_i(batch, NNODES, g);
    if (tid == 32) s_hi = lower_bound_i(batch, NNODES, g + 1);
    __syncthreads();
    const int lo = s_lo, hi = s_hi;
    const int c = tid & 63, rg = tid >> 6;
    float s = 0.0f, q = 0.0f;
    for (int n = lo + rg; n < hi; n += 4) {
        const float a = agg[(size_t)n * 64 + c];
        s += a; q += a * a;
    }
    sS[rg][c] = s; sQ[rg][c] = q;
    __syncthreads();
    if (tid < 64) {
        const float cnt = (float)max(hi - lo, 1);
        const float sum = sS[0][tid] + sS[1][tid] + sS[2][tid] + sS[3][tid];
        const float sq  = sQ[0][tid] + sQ[1][tid] + sQ[2][tid] + sQ[3][tid];
        const float m   = sum / cnt;
        const float sc  = gn_s[tid];
        float var = sq / cnt + m * m * sc * (sc - 2.0f);
        var = fmaxf(var, 0.0f);
        const float k = gn_w[tid] * rsqrtf(var + 1e-5f);
        Kb[g * 64 + tid] = k;
        Db[g * 64 + tid] = gn_b[tid] - m * sc * k;
    }
}

// ---------------------------------------------------------------------------
// Kernel 4: out = relu(agg*k + d) @ fc_W + fc_b + x   (WMMA, norm/relu fused
// into the LDS staging of the A-tile; fc_W transposed in LDS)
// ---------------------------------------------------------------------------
__global__ __launch_bounds__(128) void k_final_fc(
    const float* __restrict__ agg, const float* __restrict__ Kb,
    const float* __restrict__ Db, const int* __restrict__ batch,
    const float* __restrict__ fc_W, const float* __restrict__ fc_b,
    const float* __restrict__ x, float* __restrict__ out)
{
    __shared__ float Wt[64 * LSTR];      // fc_W transposed: [col][k]
    __shared__ float Ys[4][16 * LSTR];
    __shared__ float Bf[64];

    const int tid  = threadIdx.x;
    const int wave = tid >> 5;
    const int lane = tid & 31;
    const int half = lane >> 4;
    const int mrow = lane & 15;
    const int tile = blockIdx.x * 4 + wave;

#pragma unroll
    for (int i = 0; i < 8; ++i) {
        int f   = i * 128 + tid;
        int row = f >> 4, c4 = (f & 15) * 4;
        float4 w = *(const float4*)(fc_W + row * 64 + c4);
        Wt[(c4 + 0) * LSTR + row] = w.x;
        Wt[(c4 + 1) * LSTR + row] = w.y;
        Wt[(c4 + 2) * LSTR + row] = w.z;
        Wt[(c4 + 3) * LSTR + row] = w.w;
    }
    if (tid < 64) Bf[tid] = fc_b[tid];

    if (tile < NTILES) {
        const int base = tile * 16;
#pragma unroll
        for (int i = 0; i < 8; ++i) {
            int f   = i * 32 + lane;
            int row = f >> 4, c4 = (f & 15) * 4;
            const int n = base + row;
            const int g = batch[n];
            float4 a  = *(const float4*)(agg + (size_t)n * 64 + c4);
            float4 kk = *(const float4*)(Kb + g * 64 + c4);
            float4 dd = *(const float4*)(Db + g * 64 + c4);
            float4 y;
            y.x = fmaxf(a.x * kk.x + dd.x, 0.0f);
            y.y = fmaxf(a.y * kk.y + dd.y, 0.0f);
            y.z = fmaxf(a.z * kk.z + dd.z, 0.0f);
            y.w = fmaxf(a.w * kk.w + dd.w, 0.0f);
            *(float4*)(&Ys[wave][row * LSTR + c4]) = y;
        }
    }
    __syncthreads();
    if (tile >= NTILES) return;

    v8f acc[4] = {};
    for (int kk = 0; kk < 16; ++kk) {
        const int k = kk * 4 + half * 2;
        const v2f a = *(const v2f*)(&Ys[wave][mrow * LSTR + k]);
#pragma unroll
        for (int ct = 0; ct < 4; ++ct) {
            const int col = ct * 16 + mrow;
            const v2f b = *(const v2f*)(&Wt[col * LSTR + k]);
            acc[ct] = wmma_f32(a, b, acc[ct]);
        }
    }

    const int base = tile * 16;
#pragma unroll
    for (int r = 0; r < 8; ++r) {
        const int row = base + r + 8 * half;
#pragma unroll
        for (int ct = 0; ct < 4; ++ct) {
            const int col = ct * 16 + mrow;
            out[(size_t)row * 64 + col] =
                acc[ct][r] + Bf[col] + x[(size_t)row * 64 + col];
        }
    }
}

// ---------------------------------------------------------------------------
extern "C" void kernel_launch(void* const* d_in, const int* in_sizes, int n_in,
                              void* d_out, int out_size, void* d_ws, size_t ws_size,
                              hipStream_t stream) {
    const float* x     = (const float*)d_in[0];
    const float* pos   = (const float*)d_in[1];
    const int*   ei    = (const int*)d_in[2];
    const int*   batch = (const int*)d_in[3];
    const float* msg_W = (const float*)d_in[4];
    const float* msg_b = (const float*)d_in[5];
    const float* gn_w  = (const float*)d_in[6];
    const float* gn_b  = (const float*)d_in[7];
    const float* gn_s  = (const float*)d_in[8];
    const float* fc_W  = (const float*)d_in[9];
    const float* fc_b  = (const float*)d_in[10];
    float* out = (float*)d_out;

    char* ws = (char*)d_ws;
    const size_t NB = (size_t)NNODES * 64 * sizeof(float);   // 25.6 MB, 16B-aligned
    float* DstT = (float*)(ws);
    float* SrcT = (float*)(ws + NB);
    float* agg  = (float*)(ws + 2 * NB);
    char*  p    = ws + 3 * NB;
    float* Kb   = (float*)p;               p += 64 * 64 * sizeof(float);
    float* Db   = (float*)p;               p += 64 * 64 * sizeof(float);
    int*   Cnt  = (int*)p;                 p += (size_t)NNODES * sizeof(int);
    int*   Cur  = (int*)p;                 p += (size_t)NNODES * sizeof(int);
    int*   Off  = (int*)p;                 p += (size_t)NNODES * sizeof(int);
    int*   Bs   = (int*)p;                 p += 512 * sizeof(int);
    int*   Esrc = (int*)p;

    const int nblk  = (NNODES + 255) / 256;   // 391
    const int eblk  = (NEDGES + 255) / 256;   // 6250
    const int gemm_blocks = (NTILES + 3) / 4; // 1563

    // CSR build (by destination)
    k_zero_i<<<nblk, 256, 0, stream>>>(Cnt, Cur, NNODES);
    k_count <<<eblk, 256, 0, stream>>>(ei, Cnt);
    k_scan1 <<<nblk, 256, 0, stream>>>(Cnt, Off, Bs);
    k_scan2 <<<1,    512, 0, stream>>>(Bs);
    k_scan3 <<<nblk, 256, 0, stream>>>(Off, Bs);
    k_fill  <<<eblk, 256, 0, stream>>>(ei, Off, Cur, Esrc);

    // Node precompute (WMMA), gather-max aggregation, norm, final FC (WMMA)
    k_node_pre   <<<gemm_blocks, 128, 0, stream>>>(x, pos, msg_W, msg_b, DstT, SrcT);
    k_aggregate  <<<(NNODES + 7) / 8, 256, 0, stream>>>(DstT, SrcT, Esrc, Off, Cnt, agg);
    k_graph_stats<<<NGRAPH, 256, 0, stream>>>(agg, batch, gn_w, gn_b, gn_s, Kb, Db);
    k_final_fc   <<<gemm_blocks, 128, 0, stream>>>(agg, Kb, Db, batch, fc_W, fc_b, x, out);
}